// E3nnSeparateTSQNodeBlock_75471165325411
// MI455X (gfx1250) — compile-verified
//
#include <hip/hip_runtime.h>

typedef float v2f __attribute__((ext_vector_type(2)));
typedef float v8f __attribute__((ext_vector_type(8)));

// ---------------- problem constants ----------------
#define MUL0 32
#define MUL1 16
#define P0   496
#define P1   120
#define XROW 160
#define OUTROW 120

#define NSTEP1 128   // K1 = 512 = 16*32
#define NSTEP2 34    // K2 = 136 = 16 + 120

// LDS layout (float indices)
// W0: per (sq): [kb][lane][4]  (float4 = {tile0 j0, tile0 j1, tile1 j0, tile1 j1})
#define W0_FLOATS 21248      // per square: 664*32  (166*32*4)
#define OFF_W1    42496      // 2*W0_FLOATS
#define W1_FLOATS 8192       // 512*16, layout [kb][lane][2]
#define OFF_W2    58880      // OFF_W1 + 2*W1_FLOATS
#define W2_FLOATS 2176       // 136*16 (cols 8..15 zero-padded), layout [kb][lane][2]
#define LDS_FLOATS 63232     // OFF_W2 + 2*W2_FLOATS

// path normalization constants (INV_SQRT3 folded where the reference applies it)
#define S_UUW00  0.17677669529663688f   // sqrt(1/32)
#define S_LTV00  0.04490133f            // sqrt(1/496)
#define S_UUW110 0.14433756729740646f   // sqrt(1/16)/sqrt(3)
#define S_LTV110 0.05270462766947299f   // sqrt(1/120)/sqrt(3)
#define S_W1     0.04419417382415922f   // sqrt(3/512)/sqrt(3) = sqrt(1/512)
#define S_UUW112 0.55901699437494742f   // sqrt(5/16)
#define S_LTV112 0.20412414523193154f   // sqrt(5/120)
#define C10      0.31622776601683794f   // 1/sqrt(10)  (C112 off-diagonal)
#define C30      0.18257418583505536f   // 1/sqrt(30)  (C112 m=2)

// ---------------- compile-time triu index tables ----------------
template<int M, int P>
struct PairTab { short u[P]; short v[P]; };

template<int M, int P>
constexpr PairTab<M, P> make_tab() {
  PairTab<M, P> t{};
  int p = 0;
  for (int a = 0; a < M; ++a)
    for (int b = a + 1; b < M; ++b) { t.u[p] = (short)a; t.v[p] = (short)b; ++p; }
  return t;
}
__device__ constexpr PairTab<MUL0, P0> TAB0 = make_tab<MUL0, P0>();
__device__ constexpr PairTab<MUL1, P1> TAB1 = make_tab<MUL1, P1>();

// ---------------- WMMA helper: D(16x16,f32) = A(16x4,f32) * B(4x16,f32) + C ----------------
static __device__ __forceinline__ v8f wmma4(v2f a, v2f b, v8f c) {
  return __builtin_amdgcn_wmma_f32_16x16x4_f32(false, a, false, b, (short)0, c,
                                               false, false);
}

struct Params {
  const float* x0;
  const float* x1;
  const float* w[14];   // [sq*7 + {uuw00,ltv00,uvw101,uuw110,ltv110,uuw112,ltv112}]
  float* out;
  int n;
  int nTiles;
};

extern __shared__ float lds[];

#define VDOT(UA, UB) (varr[3*(UA)+0]*varr[3*(UB)+0] + \
                      varr[3*(UA)+1]*varr[3*(UB)+1] + \
                      varr[3*(UA)+2]*varr[3*(UB)+2])

// One out0 K-step: select this lane-half's feature pair, load both N-tiles'
// B fragments with a single ds_load_b128, advance both accumulator chains.
#define OUT0_STEP(KB, Q0, Q1, Q2, Q3)                                   \
  {                                                                     \
    v2f qlo; qlo.x = (Q0); qlo.y = (Q1);                                \
    v2f qhi; qhi.x = (Q2); qhi.y = (Q3);                                \
    const v2f a = hiH ? qhi : qlo;                                      \
    const float4 bb = B0[(KB) * 32];                                    \
    v2f b0; b0.x = bb.x; b0.y = bb.y;                                   \
    v2f b1; b1.x = bb.z; b1.y = bb.w;                                   \
    c0[0] = wmma4(a, b0, c0[0]);                                        \
    c0[1] = wmma4(a, b1, c0[1]);                                        \
  }

__global__ __launch_bounds__(256, 1)
void tsq_kernel(Params P) {
  const int tid = threadIdx.x;

  // ============ stage 0: pack all weights into LDS in B-fragment layout ============
  for (int e = tid; e < LDS_FLOATS; e += (int)blockDim.x) {
    float val;
    if (e < 2 * W0_FLOATS) {
      const int sq = e / W0_FLOATS;
      int r = e - sq * W0_FLOATS;
      const int kb = r >> 7;            // 128 floats per K-step (32 lanes x float4)
      const int r3 = r & 127;
      const int lane = r3 >> 2;
      const int q = r3 & 3;
      const int tile = q >> 1, j = q & 1;
      const int K = 4 * kb + 2 * (lane >> 4) + j;
      const int col = tile * 16 + (lane & 15);
      const float* const* w = &P.w[sq * 7];
      if (K < MUL0)                  val = w[0][K * 32 + col] * S_UUW00;
      else if (K < MUL0 + P0)        val = w[1][(K - MUL0) * 32 + col] * S_LTV00;
      else if (K < MUL0 + P0 + MUL1) val = w[3][(K - MUL0 - P0) * 32 + col] * S_UUW110;
      else                           val = w[4][(K - MUL0 - P0 - MUL1) * 32 + col] * S_LTV110;
      lds[e] = val;
    } else if (e < OFF_W2) {
      int r = e - OFF_W1;
      const int sq = r / W1_FLOATS;
      r -= sq * W1_FLOATS;
      const int kb = r >> 6, le = r & 63;
      const int lane = le >> 1, j = le & 1;
      const int K = 4 * kb + 2 * (lane >> 4) + j;       // K = u*32 + vv
      const int col = lane & 15;
      val = P.w[sq * 7 + 2][K * 16 + col] * S_W1;
      lds[e] = val;
    } else {
      int r = e - OFF_W2;
      const int sq = r / W2_FLOATS;
      r -= sq * W2_FLOATS;
      const int kb = r >> 6, le = r & 63;
      const int lane = le >> 1, j = le & 1;
      const int K = 4 * kb + 2 * (lane >> 4) + j;
      const int col = lane & 15;
      if (col >= 8)      val = 0.0f;
      else if (K < MUL1) val = P.w[sq * 7 + 5][K * 8 + col] * S_UUW112;
      else               val = P.w[sq * 7 + 6][(K - MUL1) * 8 + col] * S_LTV112;
      lds[e] = val;
    }
  }
  __syncthreads();

  // ============ stage 1: per-wave 16-node tiles ============
  const int lane  = tid & 31;
  const bool hiH  = lane >= 16;      // lane half holds K slots {2,3} of each group
  const int nrow  = lane & 15;       // node row within tile / output N column
  const int waveId = blockIdx.x * ((int)blockDim.x >> 5) + (tid >> 5);
  const int nWaves = (int)gridDim.x * ((int)blockDim.x >> 5);
  const v8f vzero = {0, 0, 0, 0, 0, 0, 0, 0};

  for (int tile = waveId; tile < P.nTiles; tile += nWaves) {
    // prefetch next tile's rows into cache while we compute this one
    const int nextTile = tile + nWaves;
    if (nextTile < P.nTiles) {
      const size_t nn = (size_t)(nextTile * 16 + nrow) * XROW;
      __builtin_prefetch(P.x0 + nn, 0, 0);
      __builtin_prefetch(P.x1 + nn, 0, 0);
    }

    v8f c0[2];
    v8f c1[3], c2[5];
    c0[0] = vzero; c0[1] = vzero;
#pragma unroll
    for (int i = 0; i < 3; ++i) c1[i] = vzero;
#pragma unroll
    for (int m = 0; m < 5; ++m) c2[m] = vzero;

#pragma clang loop unroll(disable)
    for (int sq = 0; sq < 2; ++sq) {
      int node = tile * 16 + nrow;
      if (node >= P.n) node = P.n - 1;              // branch-free clamp keeps EXEC full
      const float* xrow = (sq ? P.x1 : P.x0) + (size_t)node * XROW;

      // load this node's irreps into registers (both lane halves load same node)
      float s[32];
      float varr[48];                                // varr[3*u + i]
      const float4* xv = (const float4*)xrow;
#pragma unroll 8
      for (int q = 0; q < 8; ++q) {
        const float4 t = xv[q];
        s[4 * q + 0] = t.x; s[4 * q + 1] = t.y; s[4 * q + 2] = t.z; s[4 * q + 3] = t.w;
      }
#pragma unroll 12
      for (int q = 0; q < 12; ++q) {
        const float4 t = xv[8 + q];
        varr[4 * q + 0] = t.x; varr[4 * q + 1] = t.y; varr[4 * q + 2] = t.z; varr[4 * q + 3] = t.w;
      }

      const float4* B0 = (const float4*)(lds + sq * W0_FLOATS) + lane;
      const v2f* B1f = (const v2f*)(lds + OFF_W1 + sq * W1_FLOATS) + lane;
      const v2f* B2f = (const v2f*)(lds + OFF_W2 + sq * W2_FLOATS) + lane;

      // ---------- out0: K=664 features -> 32 cols, as 4 small homogeneous loops ----------
      // region A: s*s  (K = 0..31)
#pragma unroll 8
      for (int kb = 0; kb < 8; ++kb) {
        const int k0 = 4 * kb;
        OUT0_STEP(kb,
                  s[k0 + 0] * s[k0 + 0], s[k0 + 1] * s[k0 + 1],
                  s[k0 + 2] * s[k0 + 2], s[k0 + 3] * s[k0 + 3]);
      }
      // region B: s[iu]*s[iv]  (K = 32..527)
#pragma unroll 124
      for (int i = 0; i < 124; ++i) {
        const int p = 4 * i;
        OUT0_STEP(8 + i,
                  s[TAB0.u[p + 0]] * s[TAB0.v[p + 0]],
                  s[TAB0.u[p + 1]] * s[TAB0.v[p + 1]],
                  s[TAB0.u[p + 2]] * s[TAB0.v[p + 2]],
                  s[TAB0.u[p + 3]] * s[TAB0.v[p + 3]]);
      }
      // region C: |v_u|^2  (K = 528..543)
#pragma unroll 4
      for (int i = 0; i < 4; ++i) {
        const int u = 4 * i;
        OUT0_STEP(132 + i,
                  VDOT(u + 0, u + 0), VDOT(u + 1, u + 1),
                  VDOT(u + 2, u + 2), VDOT(u + 3, u + 3));
      }
      // region D: v_iu . v_iv  (K = 544..663)
#pragma unroll 30
      for (int i = 0; i < 30; ++i) {
        const int p = 4 * i;
        OUT0_STEP(136 + i,
                  VDOT(TAB1.u[p + 0], TAB1.v[p + 0]),
                  VDOT(TAB1.u[p + 1], TAB1.v[p + 1]),
                  VDOT(TAB1.u[p + 2], TAB1.v[p + 2]),
                  VDOT(TAB1.u[p + 3], TAB1.v[p + 3]));
      }

      // ---------- out1: K=512 features (v_ui * s_v) -> 16 cols, x3 components ----------
#pragma unroll 128
      for (int kb = 0; kb < NSTEP1; ++kb) {
        const int f0 = 4 * kb;
        const int u = f0 >> 5;
        const int vv = f0 & 31;
        v2f slo; slo.x = s[vv + 0]; slo.y = s[vv + 1];
        v2f shi; shi.x = s[vv + 2]; shi.y = s[vv + 3];
        const v2f sv = hiH ? shi : slo;
        const v2f b = B1f[kb * 32];
#pragma unroll 3
        for (int i = 0; i < 3; ++i) {
          const v2f a = sv * varr[3 * u + i];
          c1[i] = wmma4(a, b, c1[i]);
        }
      }

      // ---------- out2: K=136 features (z / zp via C112) -> 8 cols, x5 m-components ----------
#pragma unroll 34
      for (int kb = 0; kb < NSTEP2; ++kb) {
        const int k0 = 4 * kb;
        float zf[4][5];
#pragma unroll 4
        for (int t = 0; t < 4; ++t) {
          const int K = k0 + t;
          float ax, ay, az, bx, by, bz;
          if (K < MUL1) {                             // diagonal (z): a = b = v[u]
            ax = varr[3 * K + 0]; ay = varr[3 * K + 1]; az = varr[3 * K + 2];
            bx = ax; by = ay; bz = az;
          } else {                                    // pair (zp)
            const int p = K - MUL1;
            const int ua = TAB1.u[p], ub = TAB1.v[p];
            ax = varr[3 * ua + 0]; ay = varr[3 * ua + 1]; az = varr[3 * ua + 2];
            bx = varr[3 * ub + 0]; by = varr[3 * ub + 1]; bz = varr[3 * ub + 2];
          }
          const float xx = ax * bx, yy = ay * by, zz = az * bz;
          zf[t][0] = C10 * (ax * by + ay * bx);
          zf[t][1] = C10 * (ay * bz + az * by);
          zf[t][2] = C30 * (2.0f * zz - xx - yy);
          zf[t][3] = C10 * (ax * bz + az * bx);
          zf[t][4] = C10 * (xx - yy);
        }
        const v2f b = B2f[kb * 32];
#pragma unroll 5
        for (int m = 0; m < 5; ++m) {
          v2f zlo; zlo.x = zf[0][m]; zlo.y = zf[1][m];
          v2f zhi; zhi.x = zf[2][m]; zhi.y = zf[3][m];
          const v2f a = hiH ? zhi : zlo;
          c2[m] = wmma4(a, b, c2[m]);
        }
      }
    } // sq

    // ---------- store: C layout VGPR r -> row (8*half + r), col = nrow ----------
    const int rbase = tile * 16 + (hiH ? 8 : 0);
#pragma unroll 8
    for (int r = 0; r < 8; ++r) {
      const int node = rbase + r;
      if (node < P.n) {
        float* row = P.out + (size_t)node * OUTROW;
        row[nrow]              = c0[0][r];
        row[16 + nrow]         = c0[1][r];
        row[32 + 3 * nrow + 0] = c1[0][r];
        row[32 + 3 * nrow + 1] = c1[1][r];
        row[32 + 3 * nrow + 2] = c1[2][r];
        if (nrow < 8) {
          row[80 + 5 * nrow + 0] = c2[0][r];
          row[80 + 5 * nrow + 1] = c2[1][r];
          row[80 + 5 * nrow + 2] = c2[2][r];
          row[80 + 5 * nrow + 3] = c2[3][r];
          row[80 + 5 * nrow + 4] = c2[4][r];
        }
      }
    }
  } // tile loop
}

extern "C" void kernel_launch(void* const* d_in, const int* in_sizes, int n_in,
                              void* d_out, int out_size, void* d_ws, size_t ws_size,
                              hipStream_t stream) {
  (void)n_in; (void)out_size; (void)d_ws; (void)ws_size;

  Params P;
  P.x0 = (const float*)d_in[0];
  P.x1 = (const float*)d_in[1];
  for (int k = 0; k < 14; ++k) P.w[k] = (const float*)d_in[2 + k];
  P.out = (float*)d_out;
  P.n = in_sizes[0] / XROW;
  P.nTiles = (P.n + 15) / 16;

  int blocks = (P.nTiles + 15) / 16;   // ~2 tiles per wave (8 waves/block)
  if (blocks < 1) blocks = 1;
  if (blocks > 1024) blocks = 1024;

  const size_t ldsBytes = (size_t)LDS_FLOATS * sizeof(float);   // ~247 KB (<= 320 KB/WGP)
  hipFuncSetAttribute((const void*)tsq_kernel,
                      hipFuncAttributeMaxDynamicSharedMemorySize, (int)ldsBytes);
  tsq_kernel<<<dim3(blocks), dim3(256), ldsBytes, stream>>>(P);
}